// Model_69776038691104
// MI455X (gfx1250) — compile-verified
//
#include <hip/hip_runtime.h>

// MI455X / gfx1250: wave32, WMMA f32 16x16x4.
typedef __attribute__((ext_vector_type(2))) float v2f;
typedef __attribute__((ext_vector_type(8))) float v8f;

#define NQ   8192
#define NR   16384
#define KNN  8
#define WAVES_PER_BLOCK 8   // 256 threads

// Branchless compare-exchange, ascending by score.
// Lowers to v_min_num/v_max_num + v_cmp + 2x v_cndmask (no exec games).
__device__ __forceinline__ void ce(float& sa, int& ia, float& sb, int& ib) {
    const bool  sw  = sb < sa;
    const int   ilo = sw ? ib : ia;
    const int   ihi = sw ? ia : ib;
    const float slo = fminf(sa, sb);
    const float shi = fmaxf(sa, sb);
    sa = slo; sb = shi; ia = ilo; ib = ihi;
}

// Branchless min-by-score (score + carried index).
__device__ __forceinline__ void takemin(float sa, int ia, float sb, int ib,
                                        float& so, int& io) {
    const bool sw = sb < sa;
    so = fminf(sa, sb);
    io = sw ? ib : ia;
}

// Merge 8 fresh candidates (scores in c, indices hb+0..7) into the running
// sorted top-8 (s, si). Single cheap reject branch; select-only body.
__device__ __forceinline__ void topk_update(const v8f& c, int hb,
                                            float (&s)[KNN], int (&si)[KNN]) {
    const float cmin = fminf(fminf(fminf(c[0], c[1]), fminf(c[2], c[3])),
                             fminf(fminf(c[4], c[5]), fminf(c[6], c[7])));
    if (cmin < s[KNN - 1]) {
        float cs[8];
        int   ci[8];
#pragma unroll
        for (int v = 0; v < 8; ++v) { cs[v] = c[v]; ci[v] = hb + v; }

        // Optimal 19-CE sorting network for 8 (ascending).
        ce(cs[0], ci[0], cs[1], ci[1]); ce(cs[2], ci[2], cs[3], ci[3]);
        ce(cs[4], ci[4], cs[5], ci[5]); ce(cs[6], ci[6], cs[7], ci[7]);
        ce(cs[0], ci[0], cs[2], ci[2]); ce(cs[1], ci[1], cs[3], ci[3]);
        ce(cs[4], ci[4], cs[6], ci[6]); ce(cs[5], ci[5], cs[7], ci[7]);
        ce(cs[1], ci[1], cs[2], ci[2]); ce(cs[5], ci[5], cs[6], ci[6]);
        ce(cs[0], ci[0], cs[4], ci[4]); ce(cs[3], ci[3], cs[7], ci[7]);
        ce(cs[1], ci[1], cs[5], ci[5]); ce(cs[2], ci[2], cs[6], ci[6]);
        ce(cs[1], ci[1], cs[4], ci[4]); ce(cs[3], ci[3], cs[6], ci[6]);
        ce(cs[2], ci[2], cs[4], ci[4]); ce(cs[3], ci[3], cs[5], ci[5]);
        ce(cs[3], ci[3], cs[4], ci[4]);

        // Top-8 of two sorted 8-lists: m[j] = min(s[j], cs[7-j]) is the
        // multiset of the 8 smallest and forms a bitonic sequence.
        float m[8];
        int   mi[8];
#pragma unroll
        for (int j = 0; j < 8; ++j)
            takemin(s[j], si[j], cs[7 - j], ci[7 - j], m[j], mi[j]);

        // Bitonic merge of 8 (distances 4, 2, 1) -> sorted ascending.
        ce(m[0], mi[0], m[4], mi[4]); ce(m[1], mi[1], m[5], mi[5]);
        ce(m[2], mi[2], m[6], mi[6]); ce(m[3], mi[3], m[7], mi[7]);
        ce(m[0], mi[0], m[2], mi[2]); ce(m[1], mi[1], m[3], mi[3]);
        ce(m[4], mi[4], m[6], mi[6]); ce(m[5], mi[5], m[7], mi[7]);
        ce(m[0], mi[0], m[1], mi[1]); ce(m[2], mi[2], m[3], mi[3]);
        ce(m[4], mi[4], m[5], mi[5]); ce(m[6], mi[6], m[7], mi[7]);

#pragma unroll
        for (int j = 0; j < 8; ++j) { s[j] = m[j]; si[j] = mi[j]; }
    }
}

__global__ __launch_bounds__(256) void knn_topk_wmma(const float* __restrict__ q,
                                                     const float* __restrict__ r,
                                                     int* __restrict__ out)
{
    const int lane  = threadIdx.x & 31;
    const int wave  = threadIdx.x >> 5;
    const int qbase = (blockIdx.x * WAVES_PER_BLOCK + wave) * 16;
    const int col   = lane & 15;
    const bool hi   = lane >= 16;
    const int hi8   = hi ? 8 : 0;

    // ---- B matrix (4x16, queries), loop invariant ----
    // lanes 0-15 : VGPR0 = K0 (qx), VGPR1 = K1 (qy)
    // lanes 16-31: VGPR0 = K2 (qz), VGPR1 = K3 (1.0)
    const float* qp = q + (size_t)(qbase + col) * 3;
    const float qx = qp[0], qy = qp[1], qz = qp[2];
    v2f b;
    b.x = hi ? qz   : qx;
    b.y = hi ? 1.0f : qy;

    // ---- per-lane running top-8, sorted ascending ----
    float s[KNN];
    int   si[KNN];
#pragma unroll
    for (int t = 0; t < KNN; ++t) { s[t] = 3.402823466e+38f; si[t] = 0x7fffffff; }

    // 2x unrolled: 32 references per trip, two independent WMMA chains.
    for (int base = 0; base < NR; base += 32) {
        const float* rp0 = r + (size_t)(base + col) * 3;
        const float* rp1 = rp0 + 48;                       // +16 rows
        const float rx0 = rp0[0], ry0 = rp0[1], rz0 = rp0[2];
        const float rx1 = rp1[0], ry1 = rp1[1], rz1 = rp1[2];
        __builtin_prefetch(rp0 + 8 * 48, 0, 3);            // ~8 chunks ahead, WGP scope

        // ---- A matrices (16x4, reference chunks) ----
        // lanes 0-15 : VGPR0 = K0 (-2rx), VGPR1 = K1 (-2ry)
        // lanes 16-31: VGPR0 = K2 (-2rz), VGPR1 = K3 (r_sq)
        const float rsq0 = rx0 * rx0 + ry0 * ry0 + rz0 * rz0;
        const float rsq1 = rx1 * rx1 + ry1 * ry1 + rz1 * rz1;
        v2f a0, a1;
        a0.x = hi ? (-2.0f * rz0) : (-2.0f * rx0);
        a0.y = hi ? rsq0          : (-2.0f * ry0);
        a1.x = hi ? (-2.0f * rz1) : (-2.0f * rx1);
        a1.y = hi ? rsq1          : (-2.0f * ry1);

        // D[i][j] = r_sq_i - 2 * (q_j . r_i)  (squared L2 minus query-constant)
        v8f z0 = {}, z1 = {};
        const v8f c0 = __builtin_amdgcn_wmma_f32_16x16x4_f32(
                false, a0, false, b, (short)0, z0, false, false);
        const v8f c1 = __builtin_amdgcn_wmma_f32_16x16x4_f32(
                false, a1, false, b, (short)0, z1, false, false);

        topk_update(c0, base + hi8,      s, si);
        topk_update(c1, base + 16 + hi8, s, si);
    }

    // ---- merge the two half-lists for this query (lanes col and col+16) ----
    float m[16];
    int   mi[16];
#pragma unroll
    for (int t = 0; t < KNN; ++t) { m[t] = s[t]; mi[t] = si[t]; }
#pragma unroll
    for (int t = 0; t < KNN; ++t) {
        // partner's sorted list, reversed -> 16-element bitonic sequence
        m[8 + t]  = __shfl_xor(s[KNN - 1 - t], 16, 32);
        mi[8 + t] = __shfl_xor(si[KNN - 1 - t], 16, 32);
    }
    // Bitonic merge network over 16 (distances 8, 4, 2, 1), branchless.
#pragma unroll
    for (int d = 8; d >= 1; d >>= 1) {
#pragma unroll
        for (int i = 0; i < 16; ++i) {
            if ((i & d) == 0) {
                ce(m[i], mi[i], m[i + d], mi[i + d]);
            }
        }
    }

    if (!hi) {
        int* o = out + (size_t)(qbase + col) * KNN;
#pragma unroll
        for (int t = 0; t < KNN; ++t) o[t] = mi[t];
    }
}

extern "C" void kernel_launch(void* const* d_in, const int* in_sizes, int n_in,
                              void* d_out, int out_size, void* d_ws, size_t ws_size,
                              hipStream_t stream) {
    const float* q = (const float*)d_in[0];   // [8192, 3]
    const float* r = (const float*)d_in[1];   // [16384, 3]
    int* out = (int*)d_out;                   // [8192, 8] indices (int32)
    (void)in_sizes; (void)n_in; (void)out_size; (void)d_ws; (void)ws_size;

    const int blocks = NQ / (16 * WAVES_PER_BLOCK);   // 64
    knn_topk_wmma<<<blocks, 256, 0, stream>>>(q, r, out);
}